// WeightedGCNPolicyNetwork_7670811590827
// MI455X (gfx1250) — compile-verified
//
#include <hip/hip_runtime.h>
#include <math.h>

#define NNODES 100000
#define NEDGES 3200000
#define HDIM   16

typedef __attribute__((ext_vector_type(2))) float v2f;
typedef __attribute__((ext_vector_type(8))) float v8f;

// ---------------- norm precompute ----------------
__global__ void k_init(float* deg, float* scal) {
  int i = blockIdx.x * blockDim.x + threadIdx.x;
  if (i < NNODES) deg[i] = 2.0f;              // improved=True self-loop weight
  if (i == 0) { scal[0] = -__builtin_inff(); scal[1] = 0.0f; }
  if (i < HDIM) scal[2 + i] = 0.0f;           // vpool accumulators
}

__global__ void k_deg(const int* __restrict__ dst, const float* __restrict__ w,
                      float* deg) {
  int e = blockIdx.x * blockDim.x + threadIdx.x;
  if (e < NEDGES) atomicAdd(&deg[dst[e]], w[e]);
}

__global__ void k_rsqrt(float* dinv) {
  int i = blockIdx.x * blockDim.x + threadIdx.x;
  if (i < NNODES) {
    float d = dinv[i];
    dinv[i] = (d > 0.0f) ? rsqrtf(d) : 0.0f;
  }
}

// ---------------- dense GEMMs on the matrix pipe ----------------
// h = x @ W1 : N x 3 times 3 x 16, K padded to 4 with zeros. One wave per 16-row tile.
__global__ void k_gemm_xw1(const float* __restrict__ x, const float* __restrict__ W1,
                           float* __restrict__ out) {
  int wave = (blockIdx.x * blockDim.x + threadIdx.x) >> 5;
  int lane = threadIdx.x & 31;
  if (wave >= NNODES / 16) return;            // wave-uniform guard (EXEC all-ones at WMMA)
  int m  = lane & 15;
  int hi = lane >> 4;                          // 0 -> K={0,1}, 1 -> K={2,3}
  int row = wave * 16 + m;
  v2f a, b;
  a.x = (hi == 0) ? x[row * 3 + 0] : x[row * 3 + 2];
  a.y = (hi == 0) ? x[row * 3 + 1] : 0.0f;     // K=3 pad
  b.x = (hi == 0) ? W1[0 * HDIM + m] : W1[2 * HDIM + m];
  b.y = (hi == 0) ? W1[1 * HDIM + m] : 0.0f;   // K=3 pad
  v8f c = {};
  c = __builtin_amdgcn_wmma_f32_16x16x4_f32(false, a, false, b, (short)0, c, false, false);
  int rbase = wave * 16 + hi * 8;
#pragma unroll
  for (int v = 0; v < 8; ++v) out[(rbase + v) * HDIM + m] = c[v];
}

// h2pre = h @ W2 : N x 16 times 16 x 16, four K=4 WMMA steps per tile.
__global__ void k_gemm_hw2(const float* __restrict__ A, const float* __restrict__ W,
                           float* __restrict__ out) {
  int wave = (blockIdx.x * blockDim.x + threadIdx.x) >> 5;
  int lane = threadIdx.x & 31;
  if (wave >= NNODES / 16) return;
  int m  = lane & 15;
  int hi = lane >> 4;
  int row = wave * 16 + m;
  v8f c = {};
#pragma unroll
  for (int k0 = 0; k0 < HDIM; k0 += 4) {
    int kk = k0 + 2 * hi;
    v2f a, b;
    a.x = A[row * HDIM + kk];
    a.y = A[row * HDIM + kk + 1];
    b.x = W[kk * HDIM + m];
    b.y = W[(kk + 1) * HDIM + m];
    c = __builtin_amdgcn_wmma_f32_16x16x4_f32(false, a, false, b, (short)0, c, false, false);
  }
  int rbase = wave * 16 + hi * 8;
#pragma unroll
  for (int v = 0; v < 8; ++v) out[(rbase + v) * HDIM + m] = c[v];
}

// ---------------- edge aggregation (memory-bound core) ----------------
__global__ void k_self16(const float* __restrict__ hpre, const float* __restrict__ dinv,
                         float* __restrict__ out) {
  int i = blockIdx.x * blockDim.x + threadIdx.x;  // over N*16
  if (i < NNODES * HDIM) {
    float dv = dinv[i >> 4];
    out[i] = hpre[i] * (2.0f * dv * dv);          // self-loop contribution
  }
}

__global__ void k_edge16(const int* __restrict__ src, const int* __restrict__ dst,
                         const float* __restrict__ w, const float* __restrict__ dinv,
                         const float* __restrict__ hpre, float* __restrict__ out) {
  unsigned gid = blockIdx.x * blockDim.x + threadIdx.x;   // E*16 threads, 16 per edge
  int e = (int)(gid >> 4);
  int f = (int)(gid & 15);
  if (e < NEDGES) {
    int s = src[e], d = dst[e];
    float n = dinv[s] * w[e] * dinv[d];
    atomicAdd(&out[d * HDIM + f], hpre[s * HDIM + f] * n);
  }
}

__global__ void k_bias_relu16(float* __restrict__ h, const float* __restrict__ b) {
  int i = blockIdx.x * blockDim.x + threadIdx.x;
  if (i < NNODES * HDIM) {
    float v = h[i] + b[i & 15];
    h[i] = v > 0.0f ? v : 0.0f;
  }
}

// ---------------- conv3 (scalar width) ----------------
__global__ void k_dot16(const float* __restrict__ h, const float* __restrict__ W3,
                        float* __restrict__ cpre) {
  int i = blockIdx.x * blockDim.x + threadIdx.x;
  if (i < NNODES) {
    float s = 0.0f;
#pragma unroll
    for (int j = 0; j < HDIM; ++j) s += h[i * HDIM + j] * W3[j];
    cpre[i] = s;
  }
}

__global__ void k_self1(const float* __restrict__ cpre, const float* __restrict__ dinv,
                        float* __restrict__ cagg) {
  int i = blockIdx.x * blockDim.x + threadIdx.x;
  if (i < NNODES) {
    float dv = dinv[i];
    cagg[i] = cpre[i] * (2.0f * dv * dv);
  }
}

__global__ void k_edge1(const int* __restrict__ src, const int* __restrict__ dst,
                        const float* __restrict__ w, const float* __restrict__ dinv,
                        const float* __restrict__ cpre, float* __restrict__ cagg) {
  int e = blockIdx.x * blockDim.x + threadIdx.x;
  if (e < NEDGES) {
    int s = src[e], d = dst[e];
    atomicAdd(&cagg[d], cpre[s] * (dinv[s] * w[e] * dinv[d]));
  }
}

// ---------------- pooling + softmax ----------------
__global__ void k_pool(const float* __restrict__ h, float* __restrict__ scal) {
  __shared__ float sd[256];
  int tid = threadIdx.x;
  int gid = blockIdx.x * blockDim.x + tid;      // N*16 exactly covers grid
  sd[tid] = (gid < NNODES * HDIM) ? h[gid] : 0.0f;
  __syncthreads();
  for (int off = 128; off >= HDIM; off >>= 1) { // offsets multiple of 16: per-feature tree
    if (tid < off) sd[tid] += sd[tid + off];
    __syncthreads();
  }
  if (tid < HDIM) atomicAdd(&scal[2 + tid], sd[tid]);
}

__global__ void k_maskmax(const float* __restrict__ cagg, const float* __restrict__ b3,
                          const unsigned char* __restrict__ choices,
                          float* __restrict__ cfin, float* scal) {
  int i = blockIdx.x * blockDim.x + threadIdx.x;
  if (i < NNODES) {
    float cf = cagg[i] + b3[0];
    cfin[i] = cf;
    if (choices[i]) atomicMax(&scal[0], cf);
  }
}

__global__ void k_expsum(const float* __restrict__ cfin,
                         const unsigned char* __restrict__ choices,
                         float* __restrict__ evec, float* scal) {
  __shared__ float sd[256];
  int tid = threadIdx.x;
  int i = blockIdx.x * blockDim.x + tid;
  float mx = scal[0];
  float e = 0.0f;
  if (i < NNODES && choices[i]) e = __expf(cfin[i] - mx);
  if (i < NNODES) evec[i] = e;
  sd[tid] = e;
  __syncthreads();
  for (int off = 128; off >= 1; off >>= 1) {
    if (tid < off) sd[tid] += sd[tid + off];
    __syncthreads();
  }
  if (tid == 0) atomicAdd(&scal[1], sd[0]);
}

__global__ void k_choice(const float* __restrict__ evec, const float* scal,
                         float* __restrict__ out) {
  int i = blockIdx.x * blockDim.x + threadIdx.x;
  if (i < NNODES) out[i] = evec[i] / scal[1];
}

__global__ void k_value(const float* scal, const float* __restrict__ Wfc,
                        const float* __restrict__ bfc, float* __restrict__ out) {
  if (threadIdx.x == 0 && blockIdx.x == 0) {
    float v = 0.0f;
    const float invn = 1.0f / (float)NNODES;
#pragma unroll
    for (int j = 0; j < HDIM; ++j) v += scal[2 + j] * invn * Wfc[j];
    out[NNODES] = v + bfc[0];
  }
}

extern "C" void kernel_launch(void* const* d_in, const int* in_sizes, int n_in,
                              void* d_out, int out_size, void* d_ws, size_t ws_size,
                              hipStream_t stream) {
  const float* x  = (const float*)d_in[0];
  const int*   ei = (const int*)d_in[1];          // (2,E): row0=src, row1=dst
  const float* ea = (const float*)d_in[2];
  const unsigned char* choices = (const unsigned char*)d_in[3];  // jax bool = 1 byte
  const float* W1 = (const float*)d_in[4];
  const float* b1 = (const float*)d_in[5];
  const float* W2 = (const float*)d_in[6];
  const float* b2 = (const float*)d_in[7];
  const float* W3 = (const float*)d_in[8];
  const float* b3 = (const float*)d_in[9];
  const float* Wfc = (const float*)d_in[10];
  const float* bfc = (const float*)d_in[11];
  const int* src = ei;
  const int* dst = ei + NEDGES;
  float* out = (float*)d_out;

  // workspace layout (floats)
  float* ws   = (float*)d_ws;
  float* dinv = ws;                       // N   (deg then rsqrt in place)
  float* bufA = ws + NNODES;              // N*16
  float* bufB = bufA + NNODES * HDIM;     // N*16
  float* cpre = bufB + NNODES * HDIM;     // N
  float* cagg = cpre + NNODES;            // N
  float* scal = cagg + NNODES;            // [0]=max [1]=sum [2..17]=vpool

  const int T = 256;
  const int GN    = (NNODES + T - 1) / T;          // 391
  const int GE    = (NEDGES + T - 1) / T;          // 12500
  const int GN16  = (NNODES * HDIM) / T;           // 6250 (exact)
  const int GE16  = (NEDGES * HDIM) / T;           // 200000 (exact)
  const int GWMMA = ((NNODES / 16) + (T / 32) - 1) / (T / 32);  // 782

  // norm
  k_init  <<<GN, T, 0, stream>>>(dinv, scal);
  k_deg   <<<GE, T, 0, stream>>>(dst, ea, dinv);
  k_rsqrt <<<GN, T, 0, stream>>>(dinv);

  // conv1: x@W1 (WMMA) -> aggregate -> +b1, relu
  k_gemm_xw1   <<<GWMMA, T, 0, stream>>>(x, W1, bufA);
  k_self16     <<<GN16, T, 0, stream>>>(bufA, dinv, bufB);
  k_edge16     <<<GE16, T, 0, stream>>>(src, dst, ea, dinv, bufA, bufB);
  k_bias_relu16<<<GN16, T, 0, stream>>>(bufB, b1);

  // conv2: h1@W2 (WMMA) -> aggregate -> +b2, relu
  k_gemm_hw2   <<<GWMMA, T, 0, stream>>>(bufB, W2, bufA);
  k_self16     <<<GN16, T, 0, stream>>>(bufA, dinv, bufB);
  k_edge16     <<<GE16, T, 0, stream>>>(src, dst, ea, dinv, bufA, bufB);
  k_bias_relu16<<<GN16, T, 0, stream>>>(bufB, b2);          // bufB = h2

  // conv3 (width 1)
  k_dot16 <<<GN, T, 0, stream>>>(bufB, W3, cpre);
  k_self1 <<<GN, T, 0, stream>>>(cpre, dinv, cagg);
  k_edge1 <<<GE, T, 0, stream>>>(src, dst, ea, dinv, cpre, cagg);

  // mean pool of h2 (overlaps with conv3 dependency-wise, serialized on stream)
  k_pool <<<GN16, T, 0, stream>>>(bufB, scal);

  // masked softmax -> choice; pooled value
  k_maskmax<<<GN, T, 0, stream>>>(cagg, b3, choices, cpre, scal);
  k_expsum <<<GN, T, 0, stream>>>(cpre, choices, cagg, scal);
  k_choice <<<GN, T, 0, stream>>>(cagg, scal, out);
  k_value  <<<1, 32, 0, stream>>>(scal, Wfc, bfc, out);
}